// BasicMultimodalGNN_83150566851220
// MI455X (gfx1250) — compile-verified
//
#include <hip/hip_runtime.h>
#include <hip/hip_bf16.h>
#include <math.h>

typedef __attribute__((ext_vector_type(2))) float v2f;
typedef __attribute__((ext_vector_type(8))) float v8f;

#define N_NODES_C  100000
#define NODE_DIM_C 128
#define HID_C      16
#define N_GRAPHS_C 256

// ---------------------------------------------------------------------------
// Kernel 0: reset accumulators (must run every call: graph replay determinism)
//   deg[n] = 1.0 (self loop), outacc[n*16+f] = 0, pool/cnt = 0
// ---------------------------------------------------------------------------
__global__ void init_kernel(float* __restrict__ deg, float* __restrict__ outacc,
                            float* __restrict__ pool, float* __restrict__ cnt,
                            int n_nodes, int n_graphs) {
    int i = blockIdx.x * blockDim.x + threadIdx.x;
    int total = n_nodes * HID_C;
    if (i < total)            outacc[i] = 0.0f;
    if (i < n_nodes)          deg[i]    = 1.0f;
    if (i < n_graphs * HID_C) pool[i]   = 0.0f;
    if (i < n_graphs)         cnt[i]    = 0.0f;
}

// ---------------------------------------------------------------------------
// Kernel 1: in-degree accumulation over edges (self-loop already in deg init)
// ---------------------------------------------------------------------------
__global__ void deg_kernel(const int* __restrict__ dst, float* __restrict__ deg,
                           int n_edges) {
    int e = blockIdx.x * blockDim.x + threadIdx.x;
    if (e < n_edges) atomicAdd(&deg[dst[e]], 1.0f);
}

// ---------------------------------------------------------------------------
// Kernel 2: deg -> 1/sqrt(deg)   (deg >= 1 always, matches where(deg>0,...))
// ---------------------------------------------------------------------------
__global__ void dinv_kernel(float* __restrict__ deg, int n_nodes) {
    int i = blockIdx.x * blockDim.x + threadIdx.x;
    if (i < n_nodes) deg[i] = rsqrtf(deg[i]);
}

// ---------------------------------------------------------------------------
// Kernel 3: xw = x @ gcn_W   via V_WMMA_F32_16X16X4_F32
//   one wave32 computes one 16(M) x 16(N) tile, K=128 in 32 steps of 4.
//   A layout (16x4 f32): lane l holds row (l&15), K-pair base (l>>4)*2
//   B layout (4x16 f32): lane l holds col (l&15), K-pair base (l>>4)*2
//   D layout (16x16 f32): VGPR r <-> M = r + 8*(lane>=16), N = lane&15
// ---------------------------------------------------------------------------
__global__ __launch_bounds__(256) void xw_wmma_kernel(
        const float* __restrict__ x, const float* __restrict__ W,
        float* __restrict__ xw, int n_nodes) {
    int gwave = (blockIdx.x * blockDim.x + threadIdx.x) >> 5;
    int lane  = threadIdx.x & 31;
    int row0  = gwave * 16;
    if (row0 >= n_nodes) return;          // wave-uniform: EXEC all-ones inside

    int m    = lane & 15;                 // A row / B col / D col
    int koff = (lane >> 4) * 2;           // 0 or 2
    const float* xrow = x + (size_t)(row0 + m) * NODE_DIM_C;

    v8f acc = {};
#pragma unroll
    for (int k = 0; k < NODE_DIM_C; k += 4) {
        v2f a, b;
        a.x = xrow[k + koff];
        a.y = xrow[k + koff + 1];
        b.x = W[(k + koff)     * HID_C + m];
        b.y = W[(k + koff + 1) * HID_C + m];
        // 8 args: (neg_a, A, neg_b, B, c_mod, C, reuse_a, reuse_b)
        acc = __builtin_amdgcn_wmma_f32_16x16x4_f32(
                  false, a, false, b, (short)0, acc, false, false);
    }

    int mb = (lane >> 4) * 8;
#pragma unroll
    for (int r = 0; r < 8; ++r) {
        int row = row0 + mb + r;
        xw[(size_t)row * HID_C + m] = acc[r];
    }
}

// ---------------------------------------------------------------------------
// Kernel 4: edge scatter: out[dst] += xw[src] * dinv[src]*dinv[dst]
//   16 consecutive lanes handle one edge (feature f = tid&15):
//   coalesced 64B row gather + coalesced 64B row of atomics.
// ---------------------------------------------------------------------------
__global__ void scatter_kernel(const int* __restrict__ src, const int* __restrict__ dst,
                               const float* __restrict__ xw, const float* __restrict__ dinv,
                               float* __restrict__ outacc, int n_edges) {
    long long t = (long long)blockIdx.x * blockDim.x + threadIdx.x;
    int f = (int)(t & 15);
    long long e = t >> 4;
    if (e >= n_edges) return;
    int s = src[e], d = dst[e];
    float coef = dinv[s] * dinv[d];
    atomicAdd(&outacc[(size_t)d * HID_C + f], xw[(size_t)s * HID_C + f] * coef);
}

// ---------------------------------------------------------------------------
// Kernel 5: self-loop + bias + ReLU, then mean-pool accumulation by graph
// ---------------------------------------------------------------------------
__global__ void node_pool_kernel(const float* __restrict__ outacc,
                                 const float* __restrict__ xw,
                                 const float* __restrict__ dinv,
                                 const float* __restrict__ gcn_b,
                                 const int* __restrict__ batch,
                                 float* __restrict__ pool, float* __restrict__ cnt,
                                 int n_nodes) {
    long long t = (long long)blockIdx.x * blockDim.x + threadIdx.x;
    int f = (int)(t & 15);
    long long n = t >> 4;
    if (n >= n_nodes) return;
    float di  = dinv[n];
    float val = outacc[n * HID_C + f] + xw[n * HID_C + f] * di * di + gcn_b[f];
    val = fmaxf(val, 0.0f);
    int b = batch[n];
    atomicAdd(&pool[b * HID_C + f], val);
    if (f == 0) atomicAdd(&cnt[b], 1.0f);
}

// ---------------------------------------------------------------------------
// Kernel 6: four modality FCs -> mm[B,16]; thread = (graph, unit 0..15)
// ---------------------------------------------------------------------------
__global__ void mm_kernel(const float* __restrict__ mri,  const float* __restrict__ mri_W,  const float* __restrict__ mri_b,
                          const float* __restrict__ cog,  const float* __restrict__ cog_W,  const float* __restrict__ cog_b,
                          const float* __restrict__ clin, const float* __restrict__ clin_W, const float* __restrict__ clin_b,
                          const float* __restrict__ gen,  const float* __restrict__ gen_W,  const float* __restrict__ gen_b,
                          float* __restrict__ mm, int n_graphs) {
    int t = blockIdx.x * blockDim.x + threadIdx.x;
    if (t >= n_graphs * HID_C) return;
    int g = t >> 4, u = t & 15;
    const float *in, *W, *b; int dim, j;
    if (u < 4)       { in = mri  + (size_t)g * 256; W = mri_W;  b = mri_b;  dim = 256; j = u;      }
    else if (u < 8)  { in = cog  + (size_t)g * 64;  W = cog_W;  b = cog_b;  dim = 64;  j = u - 4;  }
    else if (u < 12) { in = clin + (size_t)g * 32;  W = clin_W; b = clin_b; dim = 32;  j = u - 8;  }
    else             { in = gen  + (size_t)g * 512; W = gen_W;  b = gen_b;  dim = 512; j = u - 12; }
    float s = b[j];
    for (int i = 0; i < dim; ++i) s += in[i] * W[i * 4 + j];
    mm[g * HID_C + u] = fmaxf(s, 0.0f);
}

// ---------------------------------------------------------------------------
// Kernel 7: classifier head per graph (32 lanes):
//   comb = [pool/cnt, mm] -> Linear(32,16)+ReLU -> Linear(16,3) -> log_softmax
// ---------------------------------------------------------------------------
__global__ __launch_bounds__(32) void head_kernel(
        const float* __restrict__ pool, const float* __restrict__ cnt,
        const float* __restrict__ mm,
        const float* __restrict__ W1, const float* __restrict__ b1,
        const float* __restrict__ W2, const float* __restrict__ b2,
        float* __restrict__ out) {
    __shared__ float comb[32];
    __shared__ float h[16];
    __shared__ float logit[3];
    __shared__ float lse;
    int g = blockIdx.x, l = threadIdx.x;

    if (l < 16) comb[l] = pool[g * HID_C + l] / fmaxf(cnt[g], 1.0f);
    else        comb[l] = mm[g * HID_C + (l - 16)];
    __syncthreads();

    if (l < 16) {
        float s = b1[l];
#pragma unroll
        for (int i = 0; i < 32; ++i) s += comb[i] * W1[i * 16 + l];
        h[l] = fmaxf(s, 0.0f);
    }
    __syncthreads();

    if (l < 3) {
        float s = b2[l];
#pragma unroll
        for (int i = 0; i < 16; ++i) s += h[i] * W2[i * 3 + l];
        logit[l] = s;
    }
    __syncthreads();

    if (l == 0) {
        float mx = fmaxf(logit[0], fmaxf(logit[1], logit[2]));
        float se = expf(logit[0] - mx) + expf(logit[1] - mx) + expf(logit[2] - mx);
        lse = mx + logf(se);
    }
    __syncthreads();

    if (l < 3) out[g * 3 + l] = logit[l] - lse;
}

// ---------------------------------------------------------------------------
extern "C" void kernel_launch(void* const* d_in, const int* in_sizes, int n_in,
                              void* d_out, int out_size, void* d_ws, size_t ws_size,
                              hipStream_t stream) {
    const float* x       = (const float*)d_in[0];
    const int*   eidx    = (const int*)  d_in[1];   // [2, E]
    const int*   batch   = (const int*)  d_in[2];
    const float* mri     = (const float*)d_in[3];
    const float* cog     = (const float*)d_in[4];
    const float* clin    = (const float*)d_in[5];
    const float* genet   = (const float*)d_in[6];
    const float* gcn_W   = (const float*)d_in[7];
    const float* gcn_b   = (const float*)d_in[8];
    const float* mri_W   = (const float*)d_in[9];
    const float* mri_b   = (const float*)d_in[10];
    const float* cog_W   = (const float*)d_in[11];
    const float* cog_b   = (const float*)d_in[12];
    const float* clin_W  = (const float*)d_in[13];
    const float* clin_b  = (const float*)d_in[14];
    const float* gen_W   = (const float*)d_in[15];
    const float* gen_b   = (const float*)d_in[16];
    const float* cls_W1  = (const float*)d_in[17];
    const float* cls_b1  = (const float*)d_in[18];
    const float* cls_W2  = (const float*)d_in[19];
    const float* cls_b2  = (const float*)d_in[20];
    float* out = (float*)d_out;

    const int N = in_sizes[2];            // 100000 nodes (batch length)
    const int E = in_sizes[1] / 2;        // 3200000 edges
    const int G = in_sizes[8] ? (in_sizes[3] / 256) : N_GRAPHS_C; // 256 graphs

    const int* e_src = eidx;
    const int* e_dst = eidx + E;

    // workspace carve-up (floats)
    float* xw     = (float*)d_ws;                 // N*16
    float* deg    = xw     + (size_t)N * HID_C;   // N  (becomes dinv in-place)
    float* outacc = deg    + N;                   // N*16
    float* pool   = outacc + (size_t)N * HID_C;   // G*16
    float* cnt    = pool   + (size_t)G * HID_C;   // G
    float* mm     = cnt    + G;                   // G*16

    // 0) reset accumulators
    {
        int total = N * HID_C;
        init_kernel<<<(total + 255) / 256, 256, 0, stream>>>(deg, outacc, pool, cnt, N, G);
    }
    // 1) degrees
    deg_kernel<<<(E + 255) / 256, 256, 0, stream>>>(e_dst, deg, E);
    // 2) dinv
    dinv_kernel<<<(N + 255) / 256, 256, 0, stream>>>(deg, N);
    // 3) xw = x @ W via f32 WMMA  (one wave per 16-row tile)
    {
        int tiles   = (N + 15) / 16;
        int threads = tiles * 32;
        xw_wmma_kernel<<<(threads + 255) / 256, 256, 0, stream>>>(x, gcn_W, xw, N);
    }
    // 4) edge scatter-add (16 lanes per edge)
    {
        long long t = (long long)E * HID_C;
        scatter_kernel<<<(unsigned)((t + 255) / 256), 256, 0, stream>>>(
            e_src, e_dst, xw, deg, outacc, E);
    }
    // 5) self-loop + bias + ReLU + mean-pool accumulate
    {
        long long t = (long long)N * HID_C;
        node_pool_kernel<<<(unsigned)((t + 255) / 256), 256, 0, stream>>>(
            outacc, xw, deg, gcn_b, batch, pool, cnt, N);
    }
    // 6) modality FCs
    mm_kernel<<<(G * HID_C + 255) / 256, 256, 0, stream>>>(
        mri, mri_W, mri_b, cog, cog_W, cog_b, clin, clin_W, clin_b,
        genet, gen_W, gen_b, mm, G);
    // 7) classifier head + log_softmax
    head_kernel<<<G, 32, 0, stream>>>(pool, cnt, mm, cls_W1, cls_b1, cls_W2, cls_b2, out);
}